// IOULoss_7636451852562
// MI455X (gfx1250) — compile-verified
//
#include <hip/hip_runtime.h>

// ---------------------------------------------------------------------------
// IoU loss: mean(-log(iou(preds, gts))), N = 8.4M boxes.
// Roofline: 32 B/elem * 8.39M = 268 MB @ 23.3 TB/s -> ~11.5 us floor; the
// ~40 VALU ops/elem (incl. v_log_f32) are far below VALU peak -> memory bound.
// Pass 1: grid-stride unrolled x4 -> 8 non-temporal global_load_b128 in
//         flight per wave iteration (~12+ MB aggregate MLP, saturates HBM;
//         input (268 MB) > L2 (192 MB) and is never reused, hence TH=NT).
//         Wave reduction via V_WMMA_F32_16X16X4_F32 (A = lane partials,
//         B = ones -> row sums), LDS combine, block partial -> d_ws.
// Pass 2: 1024-thread (32-wave) finisher, float4 loads of partials so the
//         reduction is wave-parallel instead of latency-serial, WMMA wave
//         reduce again, writes -sum/N.
// ---------------------------------------------------------------------------

#define BLOCK 256
#define FBLOCK 1024
#define GRID_MAX 8192

typedef __attribute__((ext_vector_type(4))) float fvec4;
typedef __attribute__((ext_vector_type(2))) float v2f;
typedef __attribute__((ext_vector_type(8))) float v8f;

// Cross-lane wave32 reduction on the matrix pipe: A(16x4) holds each lane's
// value in its first K slot, B = ones => D[m][n] = rowsum(A[m]).
// Lanes 0..15 end with rows 0-7 summed, lanes 16..31 rows 8-15; one
// half-wave shuffle completes the 32-lane sum (valid on lane 0).
// Requires EXEC all-1s: callers only invoke it uniformly from full waves.
__device__ __forceinline__ float wave_reduce_wmma(float s) {
    v2f a; a[0] = s;    a[1] = 0.0f;
    v2f b; b[0] = 1.0f; b[1] = 1.0f;
    v8f c = {};
    c = __builtin_amdgcn_wmma_f32_16x16x4_f32(
        false, a, false, b, (short)0, c, false, false);
    float rs = ((c[0] + c[1]) + (c[2] + c[3])) + ((c[4] + c[5]) + (c[6] + c[7]));
    rs += __shfl_down(rs, 16, 32);
    return rs;
}

template <int BS>
__device__ __forceinline__ float block_reduce(float v) {
    __shared__ float wsum[BS / 32];
    float w = wave_reduce_wmma(v);
    if ((threadIdx.x & 31) == 0)
        wsum[threadIdx.x >> 5] = w;
    __syncthreads();
    float t = 0.0f;
    if (threadIdx.x == 0) {
#pragma unroll
        for (int j = 0; j < BS / 32; ++j) t += wsum[j];
    }
    return t;   // valid on thread 0
}

__device__ __forceinline__ float iou_log_term(fvec4 p, fvec4 g) {
    const float EPS = 1e-7f;
    float b1x1 = __builtin_fmaf(-0.5f, p.z, p.x);
    float b1x2 = __builtin_fmaf( 0.5f, p.z, p.x);
    float b1y1 = __builtin_fmaf(-0.5f, p.w, p.y);
    float b1y2 = __builtin_fmaf( 0.5f, p.w, p.y);
    float b2x1 = __builtin_fmaf(-0.5f, g.z, g.x);
    float b2x2 = __builtin_fmaf( 0.5f, g.z, g.x);
    float b2y1 = __builtin_fmaf(-0.5f, g.w, g.y);
    float b2y2 = __builtin_fmaf( 0.5f, g.w, g.y);
    float iw = fmaxf(fminf(b1x2, b2x2) - fmaxf(b1x1, b2x1), 0.0f);
    float ih = fmaxf(fminf(b1y2, b2y2) - fmaxf(b1y1, b2y1), 0.0f);
    float inter = iw * ih;
    float uni   = p.z * p.w + g.z * g.w - inter + EPS;
    float iou   = inter / uni;
    iou = (iou == 0.0f) ? EPS : iou;   // where(iou==0, iou+eps, iou)
    return __logf(iou);
}

__global__ __launch_bounds__(BLOCK) void iou_partial_kernel(
    const fvec4* __restrict__ preds, const fvec4* __restrict__ gts,
    float* __restrict__ partials, int N)
{
    int tid    = blockIdx.x * BLOCK + threadIdx.x;
    int stride = gridDim.x * BLOCK;

    float acc = 0.0f;
    int i = tid;
    // Unrolled x4: issue 8 NT b128 loads before any consumption -> deep MLP.
    for (; i + 3 * stride < N; i += 4 * stride) {
        fvec4 p0 = __builtin_nontemporal_load(&preds[i]);
        fvec4 g0 = __builtin_nontemporal_load(&gts[i]);
        fvec4 p1 = __builtin_nontemporal_load(&preds[i + stride]);
        fvec4 g1 = __builtin_nontemporal_load(&gts[i + stride]);
        fvec4 p2 = __builtin_nontemporal_load(&preds[i + 2 * stride]);
        fvec4 g2 = __builtin_nontemporal_load(&gts[i + 2 * stride]);
        fvec4 p3 = __builtin_nontemporal_load(&preds[i + 3 * stride]);
        fvec4 g3 = __builtin_nontemporal_load(&gts[i + 3 * stride]);
        acc += iou_log_term(p0, g0);
        acc += iou_log_term(p1, g1);
        acc += iou_log_term(p2, g2);
        acc += iou_log_term(p3, g3);
    }
    for (; i < N; i += stride) {
        fvec4 p = __builtin_nontemporal_load(&preds[i]);
        fvec4 g = __builtin_nontemporal_load(&gts[i]);
        acc += iou_log_term(p, g);
    }

    float t = block_reduce<BLOCK>(acc);
    if (threadIdx.x == 0)
        partials[blockIdx.x] = t;
}

// 32-wave finisher: vectorized partial loads (wave-parallel, not
// latency-serial), WMMA wave reduction, final scale.
__global__ __launch_bounds__(FBLOCK) void iou_final_kernel(
    const float* __restrict__ partials, float* __restrict__ out,
    int numPartials, float invN)
{
    int tid = threadIdx.x;
    float s = 0.0f;

    int nv = numPartials >> 2;                   // number of float4 chunks
    const fvec4* p4 = (const fvec4*)partials;
    for (int i = tid; i < nv; i += FBLOCK) {
        fvec4 v = p4[i];
        s += (v.x + v.y) + (v.z + v.w);
    }
    for (int i = (nv << 2) + tid; i < numPartials; i += FBLOCK)
        s += partials[i];                        // tail (numPartials % 4)

    float t = block_reduce<FBLOCK>(s);
    if (tid == 0)
        out[0] = -t * invN;                      // ALPHA = 1.0
}

extern "C" void kernel_launch(void* const* d_in, const int* in_sizes, int n_in,
                              void* d_out, int out_size, void* d_ws, size_t ws_size,
                              hipStream_t stream)
{
    (void)n_in; (void)out_size; (void)ws_size;
    const fvec4* preds = (const fvec4*)d_in[0];
    const fvec4* gts   = (const fvec4*)d_in[1];
    int N = in_sizes[0] / 4;                     // rows of (N,4) f32

    float* partials = (float*)d_ws;              // GRID_MAX floats = 32 KB

    // 4 elements per thread per sweep; N = 8.39M -> exactly 8192 blocks.
    int grid = (N + BLOCK * 4 - 1) / (BLOCK * 4);
    if (grid > GRID_MAX) grid = GRID_MAX;
    if (grid < 1) grid = 1;

    iou_partial_kernel<<<grid, BLOCK, 0, stream>>>(preds, gts, partials, N);
    iou_final_kernel<<<1, FBLOCK, 0, stream>>>(partials, (float*)d_out, grid,
                                               1.0f / (float)N);
}